// DiffusionModel_5557687681067
// MI455X (gfx1250) — compile-verified
//
#include <hip/hip_runtime.h>

typedef float v2f __attribute__((ext_vector_type(2)));
typedef float v8f __attribute__((ext_vector_type(8)));

#define NQ      10
#define TSTEPS  10
#define DIM     1024
#define NDATA   1024
#define LSTR    33          // 32 + 1 padding: odd stride -> conflict-free column reads

__global__ __launch_bounds__(128) void scramble_kernel(
    const float* __restrict__ in_re, const float* __restrict__ in_im,
    const float* __restrict__ phis,  const float* __restrict__ gs,
    float* __restrict__ out)
{
    __shared__ float Pr[32 * LSTR], Pi[32 * LSTR];   // state psi as 32x32 (row = top 5 bits)
    __shared__ float Tr[32 * LSTR], Ti[32 * LSTR];   // temp = UL * P
    __shared__ float ULr[32 * LSTR], ULi[32 * LSTR]; // U_left  = M0..M4 kron
    __shared__ float URr[32 * LSTR], URi[32 * LSTR]; // U_right = M5..M9 kron
    __shared__ float gate[NQ][8];                    // per qubit: re00,im00,re01,im01,re10,im10,re11,im11
    __shared__ float red[128];

    const int tid  = threadIdx.x;
    const int b    = blockIdx.x;
    const int lane = tid & 31;
    const int wave = tid >> 5;
    const int ti   = (wave >> 1) & 1;      // tile row of this wave
    const int tj   = wave & 1;             // tile col
    const int lr   = lane & 15;
    const int kh   = (lane >> 4) << 1;     // 0 for lanes 0-15, 2 for lanes 16-31
    const int hi8  = (lane >> 4) << 3;     // 0 / 8 for C/D row layout

    // ---- load state, compute norm ----
    float ssq = 0.f;
#pragma unroll
    for (int j = 0; j < 8; ++j) {
        int e = tid + 128 * j;
        float re = in_re[b * DIM + e];
        float im = in_im[b * DIM + e];
        Pr[(e >> 5) * LSTR + (e & 31)] = re;
        Pi[(e >> 5) * LSTR + (e & 31)] = im;
        ssq += re * re + im * im;
    }
    red[tid] = ssq;
    __syncthreads();
#pragma unroll
    for (int off = 64; off > 0; off >>= 1) {
        if (tid < off) red[tid] += red[tid + off];
        __syncthreads();
    }
    float scale = 1.0f / sqrtf(red[0]);
#pragma unroll
    for (int j = 0; j < 8; ++j) {
        int e = tid + 128 * j;
        Pr[(e >> 5) * LSTR + (e & 31)] *= scale;
        Pi[(e >> 5) * LSTR + (e & 31)] *= scale;
    }

    const float inv = 0.15811388300841898f; // 1/(2*sqrt(10))

    for (int tt = 0; tt < TSTEPS; ++tt) {
        __syncthreads();  // P/UL/UR/gate from previous iteration fully consumed

        // ---- per-qubit 2x2 gates ----
        if (tid < NQ) {
            int base = b * (3 * NQ * TSTEPS) + 3 * NQ * tt;
            float a  = phis[base + tid];
            float th = phis[base + NQ + tid];
            float bb = phis[base + 2 * NQ + tid];
            float s, c;  __sincosf(0.5f * th, &s, &c);
            float p = 0.5f * (a + bb), q = 0.5f * (a - bb);
            float sp, cp, sq, cq;
            __sincosf(p, &sp, &cp);
            __sincosf(q, &sq, &cq);
            gate[tid][0] =  c * cp;  gate[tid][1] = -c * sp;   // M00 = c*cis(-(a+b)/2)
            gate[tid][2] = -s * cq;  gate[tid][3] = -s * sq;   // M01 = -s*cis((a-b)/2)
            gate[tid][4] =  s * cq;  gate[tid][5] = -s * sq;   // M10 =  s*cis(-(a-b)/2)
            gate[tid][6] =  c * cp;  gate[tid][7] =  c * sp;   // M11 =  c*cis((a+b)/2)
        }
        __syncthreads();

        // ---- Kronecker expansion: UL = M0..M4, UR = M5..M9 ----
#pragma unroll
        for (int j = 0; j < 8; ++j) {
            int e  = tid + 128 * j;
            int r  = e >> 5, rp = e & 31;
            float xr = 1.f, xi = 0.f, yr = 1.f, yi = 0.f;
#pragma unroll
            for (int qq = 0; qq < 5; ++qq) {
                int bi = (r  >> (4 - qq)) & 1;
                int bj = (rp >> (4 - qq)) & 1;
                int idx = (bi * 2 + bj) * 2;
                float gr = gate[qq][idx],     gi = gate[qq][idx + 1];
                float hr = gate[5 + qq][idx], hi = gate[5 + qq][idx + 1];
                float nxr = xr * gr - xi * gi, nxi = xr * gi + xi * gr;
                float nyr = yr * hr - yi * hi, nyi = yr * hi + yi * hr;
                xr = nxr; xi = nxi; yr = nyr; yi = nyi;
            }
            ULr[r * LSTR + rp] = xr;  ULi[r * LSTR + rp] = xi;
            URr[r * LSTR + rp] = yr;  URi[r * LSTR + rp] = yi;
        }
        __syncthreads();

        // ---- matmul 1: T = UL * P ----
        // Complex via 3-mult Karatsuba: P1=ArBr, P2=AiBi, P3=(Ar+Ai)(Br+Bi)
        // re = P1-P2, im = P3-P1-P2
        {
            v8f acc1 = {}, acc2 = {}, acc3 = {};
            int m = 16 * ti + lr;          // A row for this lane
            int n = 16 * tj + lr;          // B col for this lane
#pragma unroll
            for (int kc = 0; kc < 8; ++kc) {
                int kk = 4 * kc + kh;
                v2f ar = { ULr[m * LSTR + kk], ULr[m * LSTR + kk + 1] };
                v2f ai = { ULi[m * LSTR + kk], ULi[m * LSTR + kk + 1] };
                v2f br = { Pr[kk * LSTR + n], Pr[(kk + 1) * LSTR + n] };
                v2f bi = { Pi[kk * LSTR + n], Pi[(kk + 1) * LSTR + n] };
                v2f as = ar + ai;
                v2f bs = br + bi;
                acc1 = __builtin_amdgcn_wmma_f32_16x16x4_f32(false, ar, false, br, (short)0, acc1, false, false);
                acc2 = __builtin_amdgcn_wmma_f32_16x16x4_f32(false, ai, false, bi, (short)0, acc2, false, false);
                acc3 = __builtin_amdgcn_wmma_f32_16x16x4_f32(false, as, false, bs, (short)0, acc3, false, false);
            }
#pragma unroll
            for (int v = 0; v < 8; ++v) {
                int row = 16 * ti + v + hi8;
                int col = 16 * tj + lr;
                Tr[row * LSTR + col] = acc1[v] - acc2[v];
                Ti[row * LSTR + col] = acc3[v] - acc1[v] - acc2[v];
            }
        }
        __syncthreads();

        // ---- matmul 2: P = T * UR^T, diagonal ZZ phase fused into D-write ----
        {
            v8f acc1 = {}, acc2 = {}, acc3 = {};
            int m = 16 * ti + lr;
            int n = 16 * tj + lr;          // B[k][n] = UR[n][k] -> row-major read of UR
#pragma unroll
            for (int kc = 0; kc < 8; ++kc) {
                int kk = 4 * kc + kh;
                v2f ar = { Tr[m * LSTR + kk], Tr[m * LSTR + kk + 1] };
                v2f ai = { Ti[m * LSTR + kk], Ti[m * LSTR + kk + 1] };
                v2f br = { URr[n * LSTR + kk], URr[n * LSTR + kk + 1] };
                v2f bi = { URi[n * LSTR + kk], URi[n * LSTR + kk + 1] };
                v2f as = ar + ai;
                v2f bs = br + bi;
                acc1 = __builtin_amdgcn_wmma_f32_16x16x4_f32(false, ar, false, br, (short)0, acc1, false, false);
                acc2 = __builtin_amdgcn_wmma_f32_16x16x4_f32(false, ai, false, bi, (short)0, acc2, false, false);
                acc3 = __builtin_amdgcn_wmma_f32_16x16x4_f32(false, as, false, bs, (short)0, acc3, false, false);
            }
            float theta = gs[b * TSTEPS + tt] * inv;
#pragma unroll
            for (int v = 0; v < 8; ++v) {
                int row = 16 * ti + v + hi8;
                int col = 16 * tj + lr;
                float dre = acc1[v] - acc2[v];
                float dim = acc3[v] - acc1[v] - acc2[v];
                int   pc  = __popc(row) + __popc(col);
                float zs  = 10.f - 2.f * (float)pc;
                float pairsum = 0.5f * (zs * zs - 10.f);
                float ph = -0.5f * theta * pairsum;
                float sph, cph;  __sincosf(ph, &sph, &cph);
                Pr[row * LSTR + col] = dre * cph - dim * sph;
                Pi[row * LSTR + col] = dre * sph + dim * cph;
            }
        }
    }

    __syncthreads();
    // ---- write out: plane 0 = real, plane 1 = imag ----
#pragma unroll
    for (int j = 0; j < 8; ++j) {
        int e = tid + 128 * j;
        out[b * DIM + e]               = Pr[(e >> 5) * LSTR + (e & 31)];
        out[NDATA * DIM + b * DIM + e] = Pi[(e >> 5) * LSTR + (e & 31)];
    }
}

extern "C" void kernel_launch(void* const* d_in, const int* in_sizes, int n_in,
                              void* d_out, int out_size, void* d_ws, size_t ws_size,
                              hipStream_t stream) {
    const float* in_re = (const float*)d_in[0];
    const float* in_im = (const float*)d_in[1];
    const float* phis  = (const float*)d_in[2];
    const float* gs    = (const float*)d_in[3];
    float* out = (float*)d_out;
    scramble_kernel<<<dim3(NDATA), dim3(128), 0, stream>>>(in_re, in_im, phis, gs, out);
}